// GNNEncoder_28415503630752
// MI455X (gfx1250) — compile-verified
//
#include <hip/hip_runtime.h>
#include <hip/hip_bf16.h>
#include <math.h>

typedef float     v2f  __attribute__((ext_vector_type(2)));
typedef float     v8f  __attribute__((ext_vector_type(8)));

#define NEG_SLOPE 0.2f
#define LN_EPS    1e-5f

// ---------- helpers ----------
__device__ __forceinline__ unsigned ordEncode(float f) {
  unsigned u = __float_as_uint(f);
  return (u >> 31) ? ~u : (u | 0x80000000u);
}
__device__ __forceinline__ float ordDecode(unsigned o) {
  return __uint_as_float((o >> 31) ? (o & 0x7FFFFFFFu) : ~o);
}

__global__ void fill_f32(float* p, float v, size_t n) {
  size_t i = (size_t)blockIdx.x * blockDim.x + threadIdx.x;
  if (i < n) p[i] = v;
}
__global__ void fill_u32(unsigned* p, unsigned v, size_t n) {
  size_t i = (size_t)blockIdx.x * blockDim.x + threadIdx.x;
  if (i < n) p[i] = v;
}

// pad trailing columns of each row with zeros: src[M,Ks] -> dst[M,Kd], Kd >= Ks
__global__ void pad_cols(const float* __restrict__ src, float* __restrict__ dst,
                         int M, int Ks, int Kd) {
  int i = blockIdx.x * blockDim.x + threadIdx.x;
  if (i >= M * Kd) return;
  int r = i / Kd, c = i % Kd;
  dst[i] = (c < Ks) ? src[(size_t)r * Ks + c] : 0.0f;
}

// append zero rows: src[Ms,Ncols] -> dst[Md,Ncols], Md >= Ms
__global__ void pad_rows(const float* __restrict__ src, float* __restrict__ dst,
                         int Ms, int Md, int Ncols) {
  int i = blockIdx.x * blockDim.x + threadIdx.x;
  if (i >= Md * Ncols) return;
  int r = i / Ncols;
  dst[i] = (r < Ms) ? src[i] : 0.0f;
}

// ---------- WMMA GEMM: C[M,N] = A[M,K] @ B[K,N], row-major, f32 ----------
// K is a compile-time multiple of 4 (inputs pre-padded) -> unconditional loads,
// fully unrolled K loop, one v_wmma_f32_16x16x4_f32 per 4-deep K step.
// One wave per 16x16 C tile; 4 M-tiles per 128-thread block. M%16==0, N%16==0.
template <int K>
__global__ void wmma_gemm(const float* __restrict__ A, const float* __restrict__ B,
                          float* __restrict__ C, int M, int N) {
  const int lane  = threadIdx.x & 31;
  const int wave  = threadIdx.x >> 5;
  const int tileM = blockIdx.x * 4 + wave;
  const int m0 = tileM * 16;
  const int n0 = blockIdx.y * 16;
  if (m0 + 16 > M) return;                 // wave-uniform; EXEC stays all-1s for WMMA
  const int half = lane >> 4;              // 0: lanes 0-15, 1: lanes 16-31
  const int l16  = lane & 15;
  const float* __restrict__ arow = A + (size_t)(m0 + l16) * K + half * 2;
  const float* __restrict__ bcol = B + n0 + l16 + (size_t)(half * 2) * N;
  v8f acc = {};
#pragma unroll
  for (int k = 0; k < K; k += 4) {
    // A frag (16x4): lane<16 holds K=k,k+1 ; lane>=16 holds K=k+2,k+3  (8B-aligned pair)
    v2f a = *(const v2f*)(arow + k);
    // B frag (4x16): VGPR0 = row k (+2 for upper half), VGPR1 = row k+1 (+2)
    v2f b;
    b.x = bcol[(size_t)k * N];
    b.y = bcol[(size_t)(k + 1) * N];
    acc = __builtin_amdgcn_wmma_f32_16x16x4_f32(false, a, false, b, (short)0, acc,
                                                false, false);
  }
  // D layout: VGPR r -> row m0 + half*8 + r, col n0 + (lane&15)
  float* __restrict__ crow = C + (size_t)(m0 + half * 8) * N + n0 + l16;
#pragma unroll
  for (int r = 0; r < 8; ++r)
    crow[(size_t)r * N] = acc[r];
}

// ---------- GATv2 edge pass 1: logits + segment max ----------
// thread t = (edge e, head h). logit = att . leaky_relu(xl[dst] + xr[src] + ea*We)
// C is a multiple of 4 and rows are 128B-aligned -> float4 gathers.
__global__ void edge_logits(const int* __restrict__ ei, const float* __restrict__ ea,
                            const float* __restrict__ xl, const float* __restrict__ xr,
                            const float* __restrict__ We, const float* __restrict__ att,
                            float* __restrict__ logit, unsigned* __restrict__ mOrd,
                            int E, int H, int C) {
  const int t = blockIdx.x * blockDim.x + threadIdx.x;
  if (t >= E * H) return;
  const int e = t / H, h = t % H;
  const int s = ei[e], d = ei[E + e];
  const float eav = ea[e];
  const float4* __restrict__ xld = (const float4*)(xl + (size_t)d * H * C + h * C);
  const float4* __restrict__ xrs = (const float4*)(xr + (size_t)s * H * C + h * C);
  const float4* __restrict__ we  = (const float4*)(We + h * C);
  const float4* __restrict__ at  = (const float4*)(att + h * C);
  float acc = 0.0f;
  const int q = C >> 2;
  for (int c = 0; c < q; ++c) {
    float4 a = xld[c], b = xrs[c], w = we[c], t4 = at[c];
    float v0 = a.x + b.x + eav * w.x; v0 = (v0 > 0.f) ? v0 : NEG_SLOPE * v0;
    float v1 = a.y + b.y + eav * w.y; v1 = (v1 > 0.f) ? v1 : NEG_SLOPE * v1;
    float v2 = a.z + b.z + eav * w.z; v2 = (v2 > 0.f) ? v2 : NEG_SLOPE * v2;
    float v3 = a.w + b.w + eav * w.w; v3 = (v3 > 0.f) ? v3 : NEG_SLOPE * v3;
    acc += t4.x * v0 + t4.y * v1 + t4.z * v2 + t4.w * v3;
  }
  logit[t] = acc;
  atomicMax(&mOrd[(size_t)d * H + h], ordEncode(acc));
}

// ---------- GATv2 edge pass 2: exp(logit - max) + segment sum ----------
__global__ void edge_exp(const int* __restrict__ ei, float* __restrict__ logit,
                         const unsigned* __restrict__ mOrd, float* __restrict__ denom,
                         int E, int H) {
  const int t = blockIdx.x * blockDim.x + threadIdx.x;
  if (t >= E * H) return;
  const int e = t / H, h = t % H;
  const int d = ei[E + e];
  const float m = ordDecode(mOrd[(size_t)d * H + h]);
  const float ex = __expf(logit[t] - m);
  logit[t] = ex;                               // store expv in place
  atomicAdd(&denom[(size_t)d * H + h], ex);
}

// ---------- GATv2 edge pass 3: scatter alpha * xr[src] ----------
__global__ void edge_aggr(const int* __restrict__ ei, const float* __restrict__ expv,
                          const float* __restrict__ denom, const float* __restrict__ xr,
                          float* __restrict__ out, int E, int H, int C) {
  const int t = blockIdx.x * blockDim.x + threadIdx.x;
  if (t >= E * H) return;
  const int e = t / H, h = t % H;
  const int s = ei[e], d = ei[E + e];
  const float alpha = expv[t] / denom[(size_t)d * H + h];
  const float4* __restrict__ xrs = (const float4*)(xr + (size_t)s * H * C + h * C);
  float* __restrict__ outd = out + (size_t)d * H * C + h * C;
  const int q = C >> 2;
  for (int c = 0; c < q; ++c) {
    float4 v = xrs[c];
    atomicAdd(&outd[4 * c + 0], alpha * v.x);
    atomicAdd(&outd[4 * c + 1], alpha * v.y);
    atomicAdd(&outd[4 * c + 2], alpha * v.z);
    atomicAdd(&outd[4 * c + 3], alpha * v.w);
  }
}

// ---------- bias + LayerNorm + ELU, one wave32 per node ----------
__global__ void node_ln_elu(const float* __restrict__ in, const float* __restrict__ bias,
                            const float* __restrict__ g, const float* __restrict__ b,
                            float* __restrict__ out, int N, int F) {
  const int node = blockIdx.x * (blockDim.x >> 5) + (threadIdx.x >> 5);
  const int lane = threadIdx.x & 31;
  if (node >= N) return;
  const int per = F / 32;                      // 4 (F=128) or 2 (F=64)
  float vals[4];
  float sum = 0.0f;
  for (int i = 0; i < per; ++i) {
    const int c = lane * per + i;
    float v = in[(size_t)node * F + c] + bias[c];
    vals[i] = v;
    sum += v;
  }
  for (int off = 16; off > 0; off >>= 1) sum += __shfl_xor(sum, off, 32);
  const float mu = sum / (float)F;
  float vs = 0.0f;
  for (int i = 0; i < per; ++i) { float dlt = vals[i] - mu; vs += dlt * dlt; }
  for (int off = 16; off > 0; off >>= 1) vs += __shfl_xor(vs, off, 32);
  const float inv = rsqrtf(vs / (float)F + LN_EPS);
  for (int i = 0; i < per; ++i) {
    const int c = lane * per + i;
    float hv = (vals[i] - mu) * inv * g[c] + b[c];
    out[(size_t)node * F + c] = (hv > 0.0f) ? hv : (__expf(hv) - 1.0f);
  }
}

// ---------- host orchestration ----------
extern "C" void kernel_launch(void* const* d_in, const int* in_sizes, int n_in,
                              void* d_out, int out_size, void* d_ws, size_t ws_size,
                              hipStream_t stream) {
  const float* x      = (const float*)d_in[0];
  const float* ea     = (const float*)d_in[1];
  const float* W1l    = (const float*)d_in[2];
  const float* W1r    = (const float*)d_in[3];
  const float* W1e    = (const float*)d_in[4];
  const float* att1   = (const float*)d_in[5];
  const float* b1     = (const float*)d_in[6];
  const float* ln1_g  = (const float*)d_in[7];
  const float* ln1_b  = (const float*)d_in[8];
  const float* W2l    = (const float*)d_in[9];
  const float* W2r    = (const float*)d_in[10];
  const float* W2e    = (const float*)d_in[11];
  const float* att2   = (const float*)d_in[12];
  const float* b2     = (const float*)d_in[13];
  const float* ln2_g  = (const float*)d_in[14];
  const float* ln2_b  = (const float*)d_in[15];
  const int*   ei     = (const int*)d_in[16];
  float*       outp   = (float*)d_out;

  const int N = in_sizes[0] / 10;   // 50000
  const int E = in_sizes[1];        // 800000 (EDIM=1)
  const int HID = 32, HEADS = 4, F1 = HEADS * HID /*128*/, OUTC = 64;
  const int KP = 12;                // layer-1 K padded 10 -> 12

  // --- workspace layout (floats), with cross-layer reuse ---
  float* ws      = (float*)d_ws;
  float* xl1     = ws;                              // N*F1
  float* xr1     = xl1 + (size_t)N * F1;            // N*F1
  float* h1      = xr1 + (size_t)N * F1;            // N*F1 (accum -> post-LN h -> out2)
  float* logit1  = h1  + (size_t)N * F1;            // E*HEADS
  unsigned* m1   = (unsigned*)(logit1 + (size_t)E * HEADS);   // N*HEADS
  float* denom1  = (float*)(m1 + (size_t)N * HEADS);          // N*HEADS
  float* xpad    = denom1 + (size_t)N * HEADS;      // N*KP  (x padded 10->12)
  float* w1lp    = xpad + (size_t)N * KP;           // KP*F1
  float* w1rp    = w1lp + (size_t)KP * F1;          // KP*F1
  // layer-2 aliases (regions dead by the time they're reused)
  float* xr2     = xl1;                             // N*OUTC  (xl1 dead after pass 1)
  float* logit2  = xl1 + (size_t)N * OUTC;          // E
  unsigned* m2   = (unsigned*)(logit2 + (size_t)E); // N
  float* denom2  = (float*)((float*)m2 + (size_t)N);// N
  float* xl2     = logit1;                          // N*OUTC (== E*HEADS floats)
  float* out2    = h1;                              // N*OUTC (h1 dead after layer-2 GEMMs)

  const unsigned ORD_NEG_INF = 0x007FFFFFu;         // ordEncode(-inf)
  const int TB = 256;
  auto blocks = [](size_t n, int tb) { return (unsigned)((n + tb - 1) / tb); };

  // ===== Layer 1 =====
  pad_cols<<<blocks((size_t)N * KP, TB), TB, 0, stream>>>(x, xpad, N, 10, KP);
  pad_rows<<<blocks((size_t)KP * F1, TB), TB, 0, stream>>>(W1l, w1lp, 10, KP, F1);
  pad_rows<<<blocks((size_t)KP * F1, TB), TB, 0, stream>>>(W1r, w1rp, 10, KP, F1);
  {
    dim3 grid((N / 16 + 3) / 4, F1 / 16);
    wmma_gemm<12><<<grid, 128, 0, stream>>>(xpad, w1lp, xl1, N, F1);
    wmma_gemm<12><<<grid, 128, 0, stream>>>(xpad, w1rp, xr1, N, F1);
  }
  fill_u32<<<blocks((size_t)N * HEADS, TB), TB, 0, stream>>>(m1, ORD_NEG_INF, (size_t)N * HEADS);
  fill_f32<<<blocks((size_t)N * HEADS, TB), TB, 0, stream>>>(denom1, 0.0f, (size_t)N * HEADS);
  fill_f32<<<blocks((size_t)N * F1, TB), TB, 0, stream>>>(h1, 0.0f, (size_t)N * F1);

  edge_logits<<<blocks((size_t)E * HEADS, TB), TB, 0, stream>>>(
      ei, ea, xl1, xr1, W1e, att1, logit1, m1, E, HEADS, HID);
  edge_exp<<<blocks((size_t)E * HEADS, TB), TB, 0, stream>>>(
      ei, logit1, m1, denom1, E, HEADS);
  edge_aggr<<<blocks((size_t)E * HEADS, TB), TB, 0, stream>>>(
      ei, logit1, denom1, xr1, h1, E, HEADS, HID);
  node_ln_elu<<<blocks((size_t)N * 32, TB), TB, 0, stream>>>(
      h1, b1, ln1_g, ln1_b, h1, N, F1);

  // ===== Layer 2 =====
  {
    dim3 grid((N / 16 + 3) / 4, OUTC / 16);
    wmma_gemm<128><<<grid, 128, 0, stream>>>(h1, W2l, xl2, N, OUTC);
    wmma_gemm<128><<<grid, 128, 0, stream>>>(h1, W2r, xr2, N, OUTC);
  }
  fill_u32<<<blocks((size_t)N, TB), TB, 0, stream>>>(m2, ORD_NEG_INF, (size_t)N);
  fill_f32<<<blocks((size_t)N, TB), TB, 0, stream>>>(denom2, 0.0f, (size_t)N);
  fill_f32<<<blocks((size_t)N * OUTC, TB), TB, 0, stream>>>(out2, 0.0f, (size_t)N * OUTC);

  edge_logits<<<blocks((size_t)E, TB), TB, 0, stream>>>(
      ei, ea, xl2, xr2, W2e, att2, logit2, m2, E, 1, OUTC);
  edge_exp<<<blocks((size_t)E, TB), TB, 0, stream>>>(
      ei, logit2, m2, denom2, E, 1);
  edge_aggr<<<blocks((size_t)E, TB), TB, 0, stream>>>(
      ei, logit2, denom2, xr2, out2, E, 1, OUTC);
  node_ln_elu<<<blocks((size_t)N * 32, TB), TB, 0, stream>>>(
      out2, b2, ln2_g, ln2_b, outp, N, OUTC);
}